// EdgeEncoding_8796093022645
// MI455X (gfx1250) — compile-verified
//
#include <hip/hip_runtime.h>
#include <hip/hip_bf16.h>

// CDNA5 / gfx1250, wave32.
typedef __attribute__((ext_vector_type(2))) float v2f;
typedef __attribute__((ext_vector_type(8))) float v8f;

// ---------------------------------------------------------------------------
// Kernel 1: scores[e][d] = dot(edge_embedding[e,:], edge_vector[d,:])
// V_WMMA_F32_16X16X4_F32, M=16 edges/tile, N=16 (D=5 real cols), K=64.
// Each wave processes NT=2 row-tiles (32 edges) with independent acc chains.
//
// Layouts (ISA 7.12.2):
//   A (16x4 f32): lanes 0-15 hold M=lane, K={0,1}; lanes 16-31 hold K={2,3}.
//   B (4x16 f32): lane holds column N=lane&15; same K split by lane half.
//   D (16x16 f32): VGPR v holds (M = v + 8*(lane>=16), N = lane&15).
//
// Trick: D column n depends only on B column n (resident in lane n), and we
// never store columns n >= D -- so lanes >= D just clamp their B row to 0 and
// load garbage unconditionally. No masking, no EXEC divergence in the loop.
// ---------------------------------------------------------------------------
__global__ void edgeenc_scores_wmma(const float* __restrict__ EE,
                                    const float* __restrict__ EV,
                                    float* __restrict__ scores,
                                    int num_edges, int D)
{
    constexpr int E  = 64;   // embedding width (K)
    constexpr int NT = 2;    // row-tiles per wave

    const int lane    = threadIdx.x & 31;
    const int wave    = blockIdx.x * (blockDim.x >> 5) + (threadIdx.x >> 5);
    const int rowBase = wave * (16 * NT);
    if (rowBase >= num_edges) return;          // wave-uniform: EXEC stays full

    const int half = lane >> 4;                // K sub-pair selector
    const int lid  = lane & 15;                // A: row-in-tile, B/D: column

    // ---- Preload entire B operand into registers (edge_vector is ~1.3 KB). ----
    const int bcol = (lid < D) ? lid : 0;      // clamp: cols >= D are discarded
    const float* __restrict__ brow = EV + (size_t)bcol * E;
    v2f bfrag[E / 4];
    #pragma unroll
    for (int kc = 0; kc < E / 4; ++kc)
        bfrag[kc] = *(const v2f*)(brow + kc * 4 + half * 2);

    // ---- A row pointers (clamped in-bounds; OOB rows never stored). ----
    const float* arow[NT];
    #pragma unroll
    for (int t = 0; t < NT; ++t) {
        int r = rowBase + t * 16 + lid;
        r = (r < num_edges) ? r : (num_edges - 1);
        arow[t] = EE + (size_t)r * E;
    }

    v8f acc[NT] = {};
    #pragma unroll
    for (int kc = 0; kc < E / 4; ++kc) {
        v2f afrag[NT];
        #pragma unroll
        for (int t = 0; t < NT; ++t)
            afrag[t] = *(const v2f*)(arow[t] + kc * 4 + half * 2);
        #pragma unroll
        for (int t = 0; t < NT; ++t)
            acc[t] = __builtin_amdgcn_wmma_f32_16x16x4_f32(
                false, afrag[t], false, bfrag[kc], (short)0, acc[t], false, false);
    }

    // ---- Store the D real columns. ----
    if (lid < D) {
        #pragma unroll
        for (int t = 0; t < NT; ++t) {
            #pragma unroll
            for (int v = 0; v < 8; ++v) {
                const int r = rowBase + t * 16 + v + 8 * half;
                if (r < num_edges)
                    scores[(size_t)r * D + lid] = acc[t][v];
            }
        }
    }
}

// ---------------------------------------------------------------------------
// Kernel 2: per-path gather + masked mean.  Branch-free: invalid slots clamp
// to index 0 (in-bounds) and are zeroed via cndmask selects.  scores (10 MB)
// stays L2-resident (192 MB L2).
// ---------------------------------------------------------------------------
__global__ void edgeenc_gather(const int* __restrict__ edge_paths,
                               const float* __restrict__ scores,
                               float* __restrict__ out,
                               int P, int D)
{
    const int p = blockIdx.x * blockDim.x + threadIdx.x;
    if (p >= P) return;

    const int* __restrict__ row = edge_paths + (size_t)p * D;
    float sum = 0.0f;
    int   cnt = 0;
    #pragma unroll 8
    for (int d = 0; d < D; ++d) {
        const int  idx   = row[d];
        const bool valid = (idx >= 0);
        const int  safe  = valid ? idx : 0;
        const float s    = scores[(size_t)safe * D + d];   // unconditional load
        sum += valid ? s : 0.0f;
        cnt += valid ? 1 : 0;
    }
    out[p] = (cnt > 0) ? (sum / (float)cnt) : 0.0f;
}

// ---------------------------------------------------------------------------
// Fallback (workspace too small for scores): fused gather + dot.
// ---------------------------------------------------------------------------
__global__ void edgeenc_fused_fallback(const float* __restrict__ edge_embedding,
                                       const int*   __restrict__ edge_paths,
                                       const float* __restrict__ edge_vector,
                                       float* __restrict__ out,
                                       int P, int D, int E)
{
    const int p = blockIdx.x * blockDim.x + threadIdx.x;
    if (p >= P) return;

    float sum = 0.0f;
    int   cnt = 0;
    for (int d = 0; d < D; ++d) {
        const int idx = edge_paths[(size_t)p * D + d];
        if (idx != -1) {
            const float4* er = (const float4*)(edge_embedding + (size_t)idx * E);
            const float4* vr = (const float4*)(edge_vector    + (size_t)d   * E);
            float dot = 0.0f;
            for (int q = 0; q < E / 4; ++q) {
                const float4 a = er[q];
                const float4 b = vr[q];
                dot += a.x * b.x + a.y * b.y + a.z * b.z + a.w * b.w;
            }
            sum += dot;
            ++cnt;
        }
    }
    out[p] = (cnt > 0) ? (sum / (float)cnt) : 0.0f;
}

extern "C" void kernel_launch(void* const* d_in, const int* in_sizes, int n_in,
                              void* d_out, int out_size, void* d_ws, size_t ws_size,
                              hipStream_t stream)
{
    // setup_inputs order: x (unused), edge_embedding, edge_paths, edge_vector
    const float* edge_embedding = (const float*)d_in[1];
    const int*   edge_paths     = (const int*)  d_in[2];
    const float* edge_vector    = (const float*)d_in[3];
    float*       out            = (float*)d_out;

    const int E         = 64;
    const int D         = in_sizes[3] / E;        // 5
    const int num_edges = in_sizes[1] / E;        // 500000
    const int P         = in_sizes[2] / D;        // 1000000

    const size_t scoresBytes = (size_t)num_edges * D * sizeof(float);

    if (ws_size >= scoresBytes) {
        float* scores = (float*)d_ws;

        // Kernel 1: 32 edges per wave (2 WMMA tiles), 8 waves per block.
        const int rowsPerWave = 32;
        const int waves  = (num_edges + rowsPerWave - 1) / rowsPerWave;
        const int wavesB = 8;
        const int grid1  = (waves + wavesB - 1) / wavesB;
        edgeenc_scores_wmma<<<grid1, wavesB * 32, 0, stream>>>(
            edge_embedding, edge_vector, scores, num_edges, D);

        // Kernel 2: one thread per path.
        const int block2 = 256;
        const int grid2  = (P + block2 - 1) / block2;
        edgeenc_gather<<<grid2, block2, 0, stream>>>(edge_paths, scores, out, P, D);
    } else {
        const int block = 256;
        const int grid  = (P + block - 1) / block;
        edgeenc_fused_fallback<<<grid, block, 0, stream>>>(
            edge_embedding, edge_paths, edge_vector, out, P, D, E);
    }
}